// ResidualVectorQuantizer_83150566851376
// MI455X (gfx1250) — compile-verified
//
#include <hip/hip_runtime.h>

typedef float v2f __attribute__((ext_vector_type(2)));
typedef float v8f __attribute__((ext_vector_type(8)));

#define Q_STAGES 8
#define K_CODES  1024
#define DIM      8
#define NTOK     (16 * 16384)            // B*T tokens
#define WAVES_PER_BLOCK 4
#define BLOCK_THREADS   (WAVES_PER_BLOCK * 32)   // wave32
#define NTILES   (NTOK / 16)             // 16384 token tiles
#define NBLOCKS  (NTILES / WAVES_PER_BLOCK)

// gfx1250 async global->LDS copy (ASYNCcnt-tracked, no VGPR staging),
// guarded so missing builtins fall back to a synchronous copy.
// Probe (round 3 diagnostic): param0 = int4 addrspace(1)*, so pass
// AS-qualified int4 ext-vector pointers.
#if defined(__AMDGCN__) && \
    __has_builtin(__builtin_amdgcn_global_load_async_to_lds_b128) && \
    __has_builtin(__builtin_amdgcn_s_wait_asynccnt)
#define HAVE_ASYNC_LDS 1
typedef int v4i_async __attribute__((ext_vector_type(4)));
typedef __attribute__((address_space(1))) v4i_async GV4;
typedef __attribute__((address_space(3))) v4i_async LV4;
#else
#define HAVE_ASYNC_LDS 0
#endif

// One wave = one 16-token tile across all 8 residual-VQ stages.
// Score GEMM per stage: D[16 tok x 16 cw] = A(16x8 resid) x B(8x16 cb) + bias,
// built from two chained V_WMMA_F32_16X16X4_F32 (K=4 each).
__global__ __launch_bounds__(BLOCK_THREADS)
void rvq_kernel(const float* __restrict__ x,
                const float* __restrict__ codebooks,
                float* __restrict__ out_quant,
                float* __restrict__ out_codes,
                float* __restrict__ loss_partials)
{
    __shared__ __align__(16) float sCB[K_CODES * DIM];   // 32 KB stage codebook
    __shared__ float sBias[K_CODES];                     // -0.5*||c||^2
    __shared__ float sRes[WAVES_PER_BLOCK][16][DIM];     // residual tiles
    __shared__ float sQnt[WAVES_PER_BLOCK][16][DIM];     // quantized accum
    __shared__ int   sIdx[WAVES_PER_BLOCK][16];          // winning codes

    const int tid  = threadIdx.x;
    const int lane = tid & 31;
    const int wave = tid >> 5;
    const int m    = lane & 15;                  // token row / codeword col slot
    const int kb   = (lane < 16) ? 0 : 2;        // f32 16x4 A/B K-slot base
    const int tile = blockIdx.x * WAVES_PER_BLOCK + wave;
    const int gt0  = tile * 16;                  // first global token of tile

    // init residual = x tile, zero quantized
    for (int i = lane; i < 16 * DIM; i += 32) {
        sRes[wave][i >> 3][i & 7] = x[(size_t)gt0 * DIM + i];
        sQnt[wave][i >> 3][i & 7] = 0.0f;
    }

    float lossAcc = 0.0f;

    for (int st = 0; st < Q_STAGES; ++st) {
        __syncthreads();  // protect sCB from previous stage's readers

        // stage codebook (32 KB) -> LDS
        const float4* src = (const float4*)(codebooks + (size_t)st * K_CODES * DIM);
        float4* dst = (float4*)sCB;
#if HAVE_ASYNC_LDS
        for (int i = tid; i < (K_CODES * DIM) / 4; i += BLOCK_THREADS)
            __builtin_amdgcn_global_load_async_to_lds_b128(
                (GV4*)(src + i), (LV4*)(dst + i), 0, 0);
#else
        for (int i = tid; i < (K_CODES * DIM) / 4; i += BLOCK_THREADS) dst[i] = src[i];
#endif

        // prefetch next stage's codebook into cache while this stage computes
        // (lowers to global_prefetch_b8; 128 threads x 256B stride covers 32KB)
        if (st + 1 < Q_STAGES) {
            const char* nxt = (const char*)(codebooks + (size_t)(st + 1) * K_CODES * DIM);
            __builtin_prefetch(nxt + (size_t)tid * 256, 0, 0);
        }

#if HAVE_ASYNC_LDS
        __builtin_amdgcn_s_wait_asynccnt(0);
#endif
        __syncthreads();

        // per-codeword bias: -0.5*||c||^2
        for (int i = tid; i < K_CODES; i += BLOCK_THREADS) {
            float s = 0.0f;
            #pragma unroll
            for (int k = 0; k < DIM; ++k) { float c = sCB[i * DIM + k]; s += c * c; }
            sBias[i] = -0.5f * s;
        }
        __syncthreads();

        // A fragments (16x4 f32 layout: lanes0-15 K=0,1 ; lanes16-31 K=2,3)
        v2f a0 = { sRes[wave][m][kb],     sRes[wave][m][kb + 1] };   // dims 0..3
        v2f a1 = { sRes[wave][m][kb + 4], sRes[wave][m][kb + 5] };   // dims 4..7

        float best[8];
        int   bidx[8];
        #pragma unroll
        for (int r = 0; r < 8; ++r) { best[r] = -3.4e38f; bidx[r] = 0; }

        // sweep 1024 codewords in 16-wide column tiles; unroll x4 so four
        // independent WMMA accumulation chains are in flight and LDS loads
        // are hoisted ahead of the waits
        #pragma unroll 4
        for (int nt = 0; nt < K_CODES / 16; ++nt) {
            const int n = nt * 16 + m;           // this lane's codeword column
            v2f b0 = { sCB[n * DIM + kb],     sCB[n * DIM + kb + 1] };
            v2f b1 = { sCB[n * DIM + kb + 4], sCB[n * DIM + kb + 5] };
            const float bias = sBias[n];
            v8f c;
            #pragma unroll
            for (int r = 0; r < 8; ++r) c[r] = bias;
            c = __builtin_amdgcn_wmma_f32_16x16x4_f32(false, a0, false, b0,
                                                      (short)0, c, false, false);
            c = __builtin_amdgcn_wmma_f32_16x16x4_f32(false, a1, false, b1,
                                                      (short)0, c, false, false);
            #pragma unroll
            for (int r = 0; r < 8; ++r) {
                // branchless update (v_cmp + v_cndmask, no exec-mask branches)
                bool take = c[r] > best[r];
                best[r] = take ? c[r] : best[r];
                bidx[r] = take ? n    : bidx[r];
            }
        }

        // 16-lane argmax butterfly within each half-wave, branchless
        // tie-break: lower index wins (matches argmin-first semantics)
        #pragma unroll
        for (int off = 8; off >= 1; off >>= 1) {
            #pragma unroll
            for (int r = 0; r < 8; ++r) {
                float ov = __shfl_xor(best[r], off, 32);
                int   oi = __shfl_xor(bidx[r], off, 32);
                bool take = (ov > best[r]) | ((ov == best[r]) & (oi < bidx[r]));
                best[r] = take ? ov : best[r];
                bidx[r] = take ? oi : bidx[r];
            }
        }
        if (m == 0) {   // lane 0 -> rows 0..7, lane 16 -> rows 8..15
            #pragma unroll
            for (int r = 0; r < 8; ++r)
                sIdx[wave][(lane >> 4) * 8 + r] = bidx[r];
        }
        __syncthreads();

        // per-token gather + residual/quantized/loss update (one lane per token)
        if (lane < 16) {
            const int idx = sIdx[wave][lane];
            float ls = 0.0f;
            #pragma unroll
            for (int k = 0; k < DIM; ++k) {
                float e  = sCB[idx * DIM + k];
                float rn = sRes[wave][lane][k] - e;
                sRes[wave][lane][k] = rn;
                sQnt[wave][lane][k] += e;
                float d = rn - e;              // (new_residual - embed)
                ls += d * d;
            }
            lossAcc += ls;
            out_codes[(size_t)st * NTOK + gt0 + lane] = (float)idx;
        }
    }

    __syncthreads();
    // straight-through: quantized output == accumulated quantized value
    for (int i = lane; i < 16 * DIM; i += 32)
        out_quant[(size_t)gt0 * DIM + i] = sQnt[wave][i >> 3][i & 7];

    // wave loss reduction -> one partial per tile (deterministic second pass)
    #pragma unroll
    for (int off = 16; off >= 1; off >>= 1)
        lossAcc += __shfl_xor(lossAcc, off, 32);
    if (lane == 0) loss_partials[tile] = lossAcc;
}

// Fixed-order deterministic sum of 16384 partials -> scalar loss
__global__ __launch_bounds__(256)
void rvq_loss_reduce(const float* __restrict__ partials, float* __restrict__ out_loss)
{
    __shared__ float sred[256];
    float s = 0.0f;
    for (int i = threadIdx.x; i < NTILES; i += 256) s += partials[i];
    sred[threadIdx.x] = s;
    __syncthreads();
    for (int w = 128; w >= 1; w >>= 1) {
        if (threadIdx.x < (unsigned)w) sred[threadIdx.x] += sred[threadIdx.x + w];
        __syncthreads();
    }
    if (threadIdx.x == 0)
        *out_loss = sred[0] * (0.25f / (float)(NTOK * DIM));  // COMMITMENT_COST * mean
}

extern "C" void kernel_launch(void* const* d_in, const int* in_sizes, int n_in,
                              void* d_out, int out_size, void* d_ws, size_t ws_size,
                              hipStream_t stream) {
    const float* x  = (const float*)d_in[0];   // (16, 16384, 8) f32
    const float* cb = (const float*)d_in[1];   // (8, 1024, 8)  f32

    float* out       = (float*)d_out;
    float* out_quant = out;                               // 2,097,152 floats
    float* out_loss  = out + (size_t)NTOK * DIM;          // 1 float
    float* out_codes = out_loss + 1;                      // 2,097,152 floats
    float* partials  = (float*)d_ws;                      // NTILES floats

    rvq_kernel<<<NBLOCKS, BLOCK_THREADS, 0, stream>>>(x, cb, out_quant, out_codes, partials);
    rvq_loss_reduce<<<1, 256, 0, stream>>>(partials, out_loss);
}